// CTCLayer_32847909880226
// MI455X (gfx1250) — compile-verified
//
#include <hip/hip_runtime.h>

#ifndef __has_builtin
#define __has_builtin(x) 0
#endif

#if __has_builtin(__builtin_amdgcn_global_load_async_to_lds_b128)
#define HAVE_ASYNC_LDS 1
#else
#define HAVE_ASYNC_LDS 0
#endif

#if __has_builtin(__builtin_amdgcn_s_wait_asynccnt)
#define HAVE_WAIT_ASYNC 1
#else
#define HAVE_WAIT_ASYNC 0
#endif

#define B_     512
#define T_     512
#define C_     96
#define L_     64
#define S_     129           // 2*L + 1
#define BLANK_ 95            // C - 1
#define NEGF   (-1e30f)
#define EPSF   (1e-7f)
#define DEPTH_ 16            // LDS row ring (15 async loads in flight)

typedef int v4i_ __attribute__((ext_vector_type(4)));

// ---------- CDNA5 async global->LDS staging (ASYNCcnt-tracked) --------------
__device__ __forceinline__ void async_row16(const float* g, float* l) {
#if HAVE_ASYNC_LDS
    // global_load_async_to_lds_b128 : copies 16 bytes from global address g to LDS address l
    __builtin_amdgcn_global_load_async_to_lds_b128(
        (__attribute__((address_space(1))) v4i_*)(g),
        (__attribute__((address_space(3))) v4i_*)(l),
        /*imm offset*/0, /*cpol*/0);
#else
    float4 v = *(const float4*)g;   // synchronous fallback
    *(float4*)l = v;
#endif
}

template <int N>
__device__ __forceinline__ void wait_async() {
#if HAVE_WAIT_ASYNC
    __builtin_amdgcn_s_wait_asynccnt(N);
#else
    asm volatile("s_wait_asynccnt %0" ::"n"(N) : "memory");
#endif
    asm volatile("" ::: "memory");   // LDS was written behind the compiler's back
}

// logaddexp in fp32 log space; NEG-safe (no -inf - -inf NaNs)
__device__ __forceinline__ float lse2(float a, float b) {
    float m = fmaxf(a, b);
    float d = fminf(a, b) - m;       // <= 0
    return m + __logf(1.0f + __expf(d));
}

// One wave32 per batch element. Lane owns 5 contiguous states s = 5*lane + j.
// Alpha lives entirely in registers; 2 shfl_up per timestep; 16-deep async ring.
__global__ __launch_bounds__(32)
void ctc_alpha_kernel(const int* __restrict__ y_true,     // [B, L] int32
                      const float* __restrict__ y_pred,   // [B, T, C] fp32 (softmax)
                      float* __restrict__ loss)           // [B]
{
    const int b    = blockIdx.x;
    const int lane = threadIdx.x;
    const float* yp  = y_pred + (size_t)b * T_ * C_;
    const int*   lab = y_true + (size_t)b * L_;

    __shared__ float row[DEPTH_][C_];   // 16 x 384 B probability rows

    // ---- start streaming immediately: rows 1..15 into the ring --------------
    if (lane < 24) {
        const float* gp = yp + lane * 4;
        #pragma unroll
        for (int p = 1; p < DEPTH_; ++p)
            async_row16(gp + (size_t)p * C_, &row[p & (DEPTH_ - 1)][lane * 4]);
    }

    // ---- static per-lane state: extended labels + skip mask for s = 5*lane+j
    int   e[5];
    bool  skip[5];
    #pragma unroll
    for (int j = 0; j < 5; ++j) {
        const int s = 5 * lane + j;
        if (s >= S_) { e[j] = BLANK_; skip[j] = false; continue; }
        if (s & 1) {
            e[j]    = lab[(s - 1) >> 1];          // index < 64, safe
            skip[j] = (s >= 3) && (e[j] != lab[(s - 3) >> 1]);
        } else {
            e[j]    = BLANK_;
            skip[j] = false;
        }
    }

    // ---- t = 0 init: alpha0[0] = logp(blank), alpha0[1] = logp(l0), rest NEG
    float a[5];
    #pragma unroll
    for (int j = 0; j < 5; ++j) a[j] = NEGF;
    if (lane == 0) {
        a[0] = __logf(yp[BLANK_] + EPSF);
        a[1] = __logf(yp[e[1]]  + EPSF);
    }

    // ---- alpha recursion over t = 1 .. T-1 ----------------------------------
    for (int t = 1; t < T_; ++t) {
        const int cur = t & (DEPTH_ - 1);

        if (t + (DEPTH_ - 1) < T_) {
            // refill slot (t-1)&15 (its reads drained last step), keep 15 in flight
            if (lane < 24)
                async_row16(yp + (size_t)(t + DEPTH_ - 1) * C_ + lane * 4,
                            &row[(t + DEPTH_ - 1) & (DEPTH_ - 1)][lane * 4]);
            wait_async<DEPTH_ - 1>();   // in-order completion => row t resident
        } else {
            wait_async<0>();            // tail: drain everything
        }

        // gather log-probs of this lane's 5 extended labels from LDS
        float lp[5];
        #pragma unroll
        for (int j = 0; j < 5; ++j)
            lp[j] = __logf(row[cur][e[j]] + EPSF);

        // cross-lane tail of previous chunk (states 5*lane-1, 5*lane-2)
        float pm1 = __shfl_up(a[4], 1, 32);
        float pm2 = __shfl_up(a[3], 1, 32);
        if (lane == 0) { pm1 = NEGF; pm2 = NEGF; }

        float na[5];
        {   // j = 0
            float m = lse2(a[0], pm1);
            na[0] = (skip[0] ? lse2(m, pm2) : m) + lp[0];
        }
        {   // j = 1
            float m = lse2(a[1], a[0]);
            na[1] = (skip[1] ? lse2(m, pm1) : m) + lp[1];
        }
        #pragma unroll
        for (int j = 2; j < 5; ++j) {
            float m = lse2(a[j], a[j - 1]);
            na[j] = (skip[j] ? lse2(m, a[j - 2]) : m) + lp[j];
        }
        #pragma unroll
        for (int j = 0; j < 5; ++j) a[j] = na[j];
    }

    // ---- loss = -logaddexp(alpha[128], alpha[127]) -> lane 25 (j=3, j=2) ----
    if (lane == 25) loss[b] = -lse2(a[3], a[2]);
}

extern "C" void kernel_launch(void* const* d_in, const int* in_sizes, int n_in,
                              void* d_out, int out_size, void* d_ws, size_t ws_size,
                              hipStream_t stream) {
    (void)in_sizes; (void)n_in; (void)out_size; (void)d_ws; (void)ws_size;
    const int*   y_true = (const int*)d_in[0];    // [512, 64] int32
    const float* y_pred = (const float*)d_in[1];  // [512, 512, 96] fp32
    float*       loss   = (float*)d_out;          // [512]
    ctc_alpha_kernel<<<dim3(B_), dim3(32), 0, stream>>>(y_true, y_pred, loss);
}